// HungarianMatcher_54271206752929
// MI455X (gfx1250) — compile-verified
//
#include <hip/hip_runtime.h>
#include <math.h>

// ---------------------------------------------------------------------------
// DETR Hungarian-matcher cost matrix for MI455X (gfx1250).
//   C[b,q,t] = L1(box_q, box_t) - softmax(logits)[q, label_t] + giou_cost
// Shapes: logits [8,1024,92] f32, boxes [8,1024,4] f32,
//         labels [1600] i32, tgt_boxes [1600,4] f32 -> out [8,1024,1600] f32.
// Bandwidth-bound streaming problem (52.4 MB out @ 23.3 TB/s ~ 2.3 us floor);
// no matrix structure -> use async global->LDS staging, LDS gather, wave32
// shuffles, NT stores instead of WMMA.
// ---------------------------------------------------------------------------

namespace {
constexpr int kB = 8, kQ = 1024, kC = 92, kT = 1600;
constexpr int kBQ = kB * kQ;                 // 8192 query rows
constexpr int kQPB = 32;                     // queries per block
constexpr int kThreads = 512;                // 16 waves (wave32)
constexpr int kPStride = 96;                 // padded prob row stride (floats)
constexpr float kInvalidCost = 1000.0f;
}  // namespace

typedef int v4i __attribute__((vector_size(16)));
typedef __attribute__((address_space(1))) v4i* global_v4i_ptr;
typedef __attribute__((address_space(3))) v4i* lds_v4i_ptr;

// ---- CDNA5 async global->LDS copy (ASYNCcnt path) -------------------------
__device__ __forceinline__ void async_g2l_b128(const void* g, void* l) {
#if defined(__AMDGCN__)
#if __has_builtin(__builtin_amdgcn_global_load_async_to_lds_b128)
  // Signature (probe-confirmed via diagnostics): (AS1 v4i*, AS3 v4i*, Ii, Ii)
  __builtin_amdgcn_global_load_async_to_lds_b128(
      (global_v4i_ptr)g, (lds_v4i_ptr)l, /*imm offset*/ 0, /*cpol*/ 0);
#else
  unsigned lofs = (unsigned)(__UINTPTR_TYPE__)(__attribute__((address_space(3))) void*)l;
  asm volatile("global_load_async_to_lds_b128 %0, %1, off"
               :
               : "v"(lofs), "v"(g)
               : "memory");
#endif
#else
  (void)g; (void)l;
#endif
}

__device__ __forceinline__ void wait_asynccnt0() {
#if defined(__AMDGCN__)
#if __has_builtin(__builtin_amdgcn_s_wait_asynccnt)
  __builtin_amdgcn_s_wait_asynccnt(0);
#else
  asm volatile("s_wait_asynccnt 0" ::: "memory");
#endif
#endif
}

__global__ __launch_bounds__(kThreads) void hungarian_cost_kernel(
    const float* __restrict__ logits,     // [BQ, 92]
    const float* __restrict__ qboxes,     // [BQ, 4] xyxy
    const int* __restrict__ tlabels,      // [T] int32
    const float* __restrict__ tboxes,     // [T, 4] xyxy
    float* __restrict__ out) {            // [BQ, T]
  // LDS: softmax table + full target table (staged once, asynchronously).
  __shared__ float s_prob[kQPB * kPStride];            // 12 KB
  __shared__ alignas(16) float s_tbox[kT * 4];         // 25.6 KB
  __shared__ alignas(16) int s_tlab[kT];               // 6.4 KB
  __shared__ float s_qbox[kQPB * 4];
  __shared__ float s_qarea[kQPB];
  __shared__ int s_qvalid[kQPB];

  const int tid = threadIdx.x;
  const int qbase = blockIdx.x * kQPB;  // first global query row of this block

  // ---- Phase 0: kick off async DMA of the target table into LDS ----------
  // 1600 * 16B box records and 400 * 16B of labels; one b128 per lane-iter.
  for (int i = tid; i < kT; i += kThreads) {
    async_g2l_b128(tboxes + 4 * i, &s_tbox[4 * i]);
  }
  for (int i = tid; i < kT / 4; i += kThreads) {
    async_g2l_b128(tlabels + 4 * i, &s_tlab[4 * i]);
  }

  // ---- Phase A: softmax for this block's 32 queries (overlaps the DMA) ---
  // 512 threads = 32 rows x 16 lanes/row; reduce across 16 lanes via shfl.
  {
    const int r = tid >> 4;       // local query row 0..31
    const int j = tid & 15;       // lane within row
    const int gq = qbase + r;
    const float* lrow = logits + (long long)gq * kC;

    float v[6];
    float m = -3.402823466e38f;
#pragma unroll
    for (int k = 0; k < 6; ++k) {
      const int c = j + 16 * k;
      const float x = (c < kC) ? lrow[c] : -3.402823466e38f;
      v[k] = x;
      m = fmaxf(m, x);
    }
#pragma unroll
    for (int off = 1; off < 16; off <<= 1) m = fmaxf(m, __shfl_xor(m, off, 32));

    float s = 0.0f;
#pragma unroll
    for (int k = 0; k < 6; ++k) {
      const int c = j + 16 * k;
      const float e = (c < kC) ? __expf(v[k] - m) : 0.0f;
      v[k] = e;
      s += e;
    }
#pragma unroll
    for (int off = 1; off < 16; off <<= 1) s += __shfl_xor(s, off, 32);

    const float inv = 1.0f / s;
#pragma unroll
    for (int k = 0; k < 6; ++k) {
      const int c = j + 16 * k;
      if (c < kC) s_prob[r * kPStride + c] = v[k] * inv;
    }

    if (j == 0) {
      const float4 qb = reinterpret_cast<const float4*>(qboxes)[gq];
      s_qbox[4 * r + 0] = qb.x;
      s_qbox[4 * r + 1] = qb.y;
      s_qbox[4 * r + 2] = qb.z;
      s_qbox[4 * r + 3] = qb.w;
      s_qarea[r] = (qb.z - qb.x) * (qb.w - qb.y);
      s_qvalid[r] = (qb.z >= qb.x) && (qb.w >= qb.y);
    }
  }

  // Drain this wave's async copies, then make all LDS visible block-wide.
  wait_asynccnt0();
  __syncthreads();

  // ---- Phase B: 16 waves x 2 queries each; lanes stride over targets -----
  const int wave = tid >> 5;
  const int lane = tid & 31;
  const int q0 = wave * 2;

  float qx[2], qy[2], qX[2], qY[2], qa[2];
  int qv[2];
#pragma unroll
  for (int i = 0; i < 2; ++i) {
    const int r = q0 + i;
    qx[i] = s_qbox[4 * r + 0];
    qy[i] = s_qbox[4 * r + 1];
    qX[i] = s_qbox[4 * r + 2];
    qY[i] = s_qbox[4 * r + 3];
    qa[i] = s_qarea[r];
    qv[i] = s_qvalid[r];
  }

  const long long outBase = (long long)qbase * kT;
  for (int t = lane; t < kT; t += 32) {
    const float4 tb = *reinterpret_cast<const float4*>(&s_tbox[4 * t]);
    const int c = s_tlab[t];
    const float ta = (tb.z - tb.x) * (tb.w - tb.y);
    const bool tval = (tb.z >= tb.x) && (tb.w >= tb.y);

#pragma unroll
    for (int i = 0; i < 2; ++i) {
      const float p = s_prob[(q0 + i) * kPStride + c];  // class-cost gather
      const float l1 = fabsf(qx[i] - tb.x) + fabsf(qy[i] - tb.y) +
                       fabsf(qX[i] - tb.z) + fabsf(qY[i] - tb.w);
      // intersection
      const float iw = fmaxf(fminf(qX[i], tb.z) - fmaxf(qx[i], tb.x), 0.0f);
      const float ih = fmaxf(fminf(qY[i], tb.w) - fmaxf(qy[i], tb.y), 0.0f);
      const float inter = iw * ih;
      const float uni = qa[i] + ta - inter;
      // enclosing box
      const float ew = fmaxf(qX[i], tb.z) - fminf(qx[i], tb.x);
      const float eh = fmaxf(qY[i], tb.w) - fminf(qy[i], tb.y);
      const float ae = fmaxf(ew, 0.0f) * fmaxf(eh, 0.0f);
      // giou = iou - (ae - uni)/ae ; fast v_rcp_f32 (cost matrix tolerance)
      const float giou = inter * __builtin_amdgcn_rcpf(uni) -
                         (ae - uni) * __builtin_amdgcn_rcpf(ae);
      const float cg = (qv[i] && tval) ? -giou : kInvalidCost;
      const float cost = l1 - p + cg;
      __builtin_nontemporal_store(cost,
                                  out + outBase + (long long)(q0 + i) * kT + t);
    }
  }
}

extern "C" void kernel_launch(void* const* d_in, const int* in_sizes, int n_in,
                              void* d_out, int out_size, void* d_ws,
                              size_t ws_size, hipStream_t stream) {
  (void)in_sizes; (void)n_in; (void)out_size; (void)d_ws; (void)ws_size;
  const float* logits = (const float*)d_in[0];   // [8,1024,92] f32
  const float* qboxes = (const float*)d_in[1];   // [8,1024,4]  f32
  const int* tlabels  = (const int*)d_in[2];     // [1600]      i32
  const float* tboxes = (const float*)d_in[3];   // [1600,4]    f32
  float* out = (float*)d_out;                    // [8,1024,1600] f32

  dim3 grid(kBQ / kQPB);   // 256 blocks
  dim3 block(kThreads);    // 512 threads = 16 wave32
  hipLaunchKernelGGL(hungarian_cost_kernel, grid, block, 0, stream,
                     logits, qboxes, tlabels, tboxes, out);
}